// LDLLoss_64536178589813
// MI455X (gfx1250) — compile-verified
//
#include <hip/hip_runtime.h>
#include <cstdint>

// Problem constants (fixed by reference): [16, 3, 512, 512] fp32
#define BATCH   16
#define HEIGHT  512
#define WIDTH   512
#define HW      (HEIGHT * WIDTH)          // 262144 = 2^18
#define CHW     (3 * HW)                  // 786432
#define NPIX    (BATCH * HW)              // 4194304
#define NELEM   (BATCH * 3 * HW)          // 12582912

// Kernel 1 geometry: 256 threads x 4 consecutive pixels (one float4) each
#define K1_BLK  256
#define K1_PXB  (K1_BLK * 4)              // 1024 px / block
#define K1_NBLK (NPIX / K1_PXB)           // 4096 blocks
#define BLKS_PER_IMG (HW / K1_PXB)        // 256 blocks / image

// Kernel 3 geometry: 32x32 output tile, 7x7 window -> 38x38 halo tile
#define TDIM    32
#define PADR    3
#define HTILE   (TDIM + 2 * PADR)         // 38
#define TSTRIDE 40                        // LDS row stride (bank-conflict pad)
#define NT_X    (WIDTH  / TDIM)           // 16
#define NT_Y    (HEIGHT / TDIM)           // 16
#define NLOSSBLK (BATCH * NT_X * NT_Y)    // 4096

typedef float v4f __attribute__((ext_vector_type(4)));

__device__ __forceinline__ int reflect512(int c) {
    // jnp.pad mode='reflect' (edge not repeated): -1->1, 512->510
    c = (c < 0) ? -c : c;
    c = (c >= 512) ? (1022 - c) : c;
    return c;
}

// CDNA5 async global->LDS copy (GLOBAL_LOAD_ASYNC_TO_LDS_B32, ASYNCcnt-tracked).
// VDST = VGPR holding the wave-relative LDS byte address, VADDR = 64-bit global
// address (GV mode, SADDR=off).
__device__ __forceinline__ void async_load_f32_to_lds(const float* gaddr, unsigned lds_off) {
    asm volatile("global_load_async_to_lds_b32 %0, %1, off"
                 :: "v"(lds_off), "v"((unsigned long long)(uintptr_t)gaddr)
                 : "memory");
}

// ---------------- Kernel 1: residuals + per-block var partials ----------------
// Dominant-traffic kernel (151 MB in): b128 loads, NT hint so the one-shot
// inputs don't evict R/E (33 MB, re-read by kernel 3) from the 192 MB L2.
__global__ void residual_kernel(const float* __restrict__ outp,
                                const float* __restrict__ ema,
                                const float* __restrict__ gt,
                                float* __restrict__ R,
                                float* __restrict__ E,
                                float* __restrict__ pvar) {
    __shared__ float reds[K1_BLK];
    __shared__ float redss[K1_BLK];
    const int tid = threadIdx.x;
    const int b   = blockIdx.x >> 8;                       // / BLKS_PER_IMG
    const int q   = ((blockIdx.x & 255) << 10) + (tid << 2);
    const int p0  = (b << 18) + q;                         // global pixel index
    const size_t i0 = (size_t)b * CHW + (size_t)q;

    const v4f g0 = __builtin_nontemporal_load((const v4f*)(gt + i0));
    const v4f g1 = __builtin_nontemporal_load((const v4f*)(gt + i0 + HW));
    const v4f g2 = __builtin_nontemporal_load((const v4f*)(gt + i0 + 2 * HW));
    const v4f o0 = __builtin_nontemporal_load((const v4f*)(outp + i0));
    const v4f o1 = __builtin_nontemporal_load((const v4f*)(outp + i0 + HW));
    const v4f o2 = __builtin_nontemporal_load((const v4f*)(outp + i0 + 2 * HW));
    const v4f e0 = __builtin_nontemporal_load((const v4f*)(ema + i0));
    const v4f e1 = __builtin_nontemporal_load((const v4f*)(ema + i0 + HW));
    const v4f e2 = __builtin_nontemporal_load((const v4f*)(ema + i0 + 2 * HW));

    const v4f rsr = __builtin_elementwise_abs(g0 - o0) +
                    __builtin_elementwise_abs(g1 - o1) +
                    __builtin_elementwise_abs(g2 - o2);
    const v4f rem = __builtin_elementwise_abs(g0 - e0) +
                    __builtin_elementwise_abs(g1 - e1) +
                    __builtin_elementwise_abs(g2 - e2);

    *(v4f*)(R + p0) = rsr;
    *(v4f*)(E + p0) = rem;

    const float s  = (rsr.x + rsr.y) + (rsr.z + rsr.w);
    const float ss = (rsr.x * rsr.x + rsr.y * rsr.y) +
                     (rsr.z * rsr.z + rsr.w * rsr.w);

    reds[tid]  = s;
    redss[tid] = ss;
    __syncthreads();
    for (int off = K1_BLK / 2; off > 0; off >>= 1) {
        if (tid < off) {
            reds[tid]  += reds[tid + off];
            redss[tid] += redss[tid + off];
        }
        __syncthreads();
    }
    if (tid == 0) {
        pvar[2 * blockIdx.x]     = reds[0];
        pvar[2 * blockIdx.x + 1] = redss[0];
    }
}

// ---------------- Kernel 2: per-image patch weight = var^0.2 ----------------
__global__ void patchw_kernel(const float* __restrict__ pvar,
                              float* __restrict__ pw) {
    __shared__ float rs[256];
    __shared__ float rss[256];
    const int b = blockIdx.x;
    const int tid = threadIdx.x;
    const int idx = b * BLKS_PER_IMG + tid;   // BLKS_PER_IMG == 256
    rs[tid]  = pvar[2 * idx];
    rss[tid] = pvar[2 * idx + 1];
    __syncthreads();
    for (int off = 128; off > 0; off >>= 1) {
        if (tid < off) {
            rs[tid]  += rs[tid + off];
            rss[tid] += rss[tid + off];
        }
        __syncthreads();
    }
    if (tid == 0) {
        const float n = (float)HW;            // var over (1,2,3) of [B,1,H,W]
        const float s = rs[0], ss = rss[0];
        const float var = (ss - s * s / n) / (n - 1.0f);
        pw[b] = powf(fmaxf(var, 0.0f), 0.2f);
    }
}

// ---------------- Kernel 3: local 7x7 variance + masked weighted loss ----------------
__global__ void loss_kernel(const float* __restrict__ R,
                            const float* __restrict__ E,
                            const float* __restrict__ pw,
                            float* __restrict__ ploss) {
    __shared__ float tR[HTILE * TSTRIDE];     // 38x(40) raw residual halo tile
    __shared__ float tHs[HTILE * TDIM];       // horizontal 7-sums of r
    __shared__ float tHss[HTILE * TDIM];      // horizontal 7-sums of r^2
    __shared__ float red[256];

    const int tid = threadIdx.x;
    const int b  = blockIdx.z;
    const int x0 = blockIdx.x * TDIM;
    const int y0 = blockIdx.y * TDIM;
    const float* Rb = R + b * HW;
    const unsigned ldsR0 = (unsigned)(uintptr_t)&tR[0];   // LDS byte base of tR

    // Stage 38x38 halo tile into LDS via CDNA5 async-to-LDS copies.
    // Reflect padding is folded into the source address, so every element
    // takes the async path.
    for (int idx = tid; idx < HTILE * HTILE; idx += 256) {
        const int ly = idx / HTILE;
        const int lx = idx - ly * HTILE;
        const int gy = reflect512(y0 + ly - PADR);
        const int gx = reflect512(x0 + lx - PADR);
        async_load_f32_to_lds(Rb + gy * WIDTH + gx,
                              ldsR0 + (unsigned)(ly * TSTRIDE + lx) * 4u);
    }
    __builtin_amdgcn_s_wait_asynccnt(0);
    __syncthreads();

    // Horizontal 7-tap sums (separable box filter), 38 rows x 32 cols.
    for (int idx = tid; idx < HTILE * TDIM; idx += 256) {
        const int row = idx >> 5;             // / TDIM
        const int col = idx & (TDIM - 1);
        const float* rp = &tR[row * TSTRIDE + col];
        float s = 0.0f, ss = 0.0f;
#pragma unroll
        for (int d = 0; d < 7; ++d) {
            const float v = rp[d];
            s += v;
            ss += v * v;
        }
        tHs[idx]  = s;
        tHss[idx] = ss;
    }
    __syncthreads();

    const float pwb = pw[b];
    float acc = 0.0f;
#pragma unroll
    for (int k = 0; k < 4; ++k) {
        const int lin = tid + k * 256;
        const int oy = lin >> 5;
        const int ox = lin & 31;
        float s = 0.0f, ss = 0.0f;
#pragma unroll
        for (int d = 0; d < 7; ++d) {
            s  += tHs[(oy + d) * TDIM + ox];
            ss += tHss[(oy + d) * TDIM + ox];
        }
        // unbiased 7x7 variance: (ss - s^2/49)/48
        const float var = (ss - s * s * (1.0f / 49.0f)) * (1.0f / 48.0f);
        const float rc = tR[(oy + PADR) * TSTRIDE + ox + PADR];      // center r_sr
        const float ec = E[b * HW + (y0 + oy) * WIDTH + (x0 + ox)];  // center r_ema
        const float w = (rc < ec) ? 0.0f : pwb * var;
        // mean(|w*out - w*gt|) over C collapses to |w| * r_sr per pixel
        acc += fabsf(w) * rc;
    }
    red[tid] = acc;
    __syncthreads();
    for (int off = 128; off > 0; off >>= 1) {
        if (tid < off) red[tid] += red[tid + off];
        __syncthreads();
    }
    if (tid == 0)
        ploss[(b * NT_Y + blockIdx.y) * NT_X + blockIdx.x] = red[0];
}

// ---------------- Kernel 4: final reduction ----------------
__global__ void finalize_kernel(const float* __restrict__ ploss,
                                float* __restrict__ out) {
    __shared__ float red[256];
    const int tid = threadIdx.x;
    float s = 0.0f;
    for (int i = tid; i < NLOSSBLK; i += 256) s += ploss[i];
    red[tid] = s;
    __syncthreads();
    for (int off = 128; off > 0; off >>= 1) {
        if (tid < off) red[tid] += red[tid + off];
        __syncthreads();
    }
    if (tid == 0) out[0] = red[0] * (1.0f / (float)NELEM);
}

extern "C" void kernel_launch(void* const* d_in, const int* in_sizes, int n_in,
                              void* d_out, int out_size, void* d_ws, size_t ws_size,
                              hipStream_t stream) {
    const float* outp = (const float*)d_in[0];   // 'output'
    const float* ema  = (const float*)d_in[1];   // 'output_ema'
    const float* gt   = (const float*)d_in[2];   // 'gt'

    float* ws    = (float*)d_ws;
    float* R     = ws;                      // NPIX floats
    float* E     = ws + NPIX;               // NPIX floats
    float* pvar  = ws + 2 * (size_t)NPIX;   // 2*K1_NBLK floats
    float* pw    = pvar + 2 * K1_NBLK;      // BATCH floats
    float* ploss = pw + BATCH;              // NLOSSBLK floats

    residual_kernel<<<K1_NBLK, K1_BLK, 0, stream>>>(outp, ema, gt, R, E, pvar);
    patchw_kernel<<<BATCH, 256, 0, stream>>>(pvar, pw);
    loss_kernel<<<dim3(NT_X, NT_Y, BATCH), 256, 0, stream>>>(R, E, pw, ploss);
    finalize_kernel<<<1, 256, 0, stream>>>(ploss, (float*)d_out);
}